// SNN_test_34342558499578
// MI455X (gfx1250) — compile-verified
//
#include <hip/hip_runtime.h>

// ALIF recurrence, T timesteps, 1 neuron.
//   i <- 0.7*i + W*x_t        (linear scan  -> WMMA Toeplitz matmul, exact)
//   v <- 0.6*v*(1-s) + i      (contraction  -> chunked w/ 96-step warm-up)
//   t <- 0.6*t + 0.1*s
//   s <- heaviside(v - (1 + t))
// d_out = [ outputs[T] | states[T][4] ]  (states row = i,v,thr,s)

typedef __attribute__((ext_vector_type(2))) float v2f;
typedef __attribute__((ext_vector_type(8))) float v8f;

// async-to-LDS builtin operand types (signature confirmed by toolchain
// diagnostic: addrspace(1)/addrspace(3) pointers to vector_size(16) int)
typedef int v4i_ __attribute__((vector_size(16)));
typedef __attribute__((address_space(1))) v4i_* gv4p;
typedef __attribute__((address_space(3))) v4i_* lv4p;

#define LEAK_V 0.6f
#define LEAK_T 0.6f
#define BASE_T 1.0f
#define ADD_T  0.1f
#define WARMUP 96
#define LOG2_07 (-0.5145731728297583f)
#define Q16 0.0033232930569601f /* 0.7^16  */
#define P8 0.05764801f          /* 0.7^8   */
#define I1 1.4285714285714286f  /* 0.7^-1  */
#define I2 2.0408163265306123f  /* 0.7^-2  */
#define I4 4.1649312786339027f  /* 0.7^-4  */
#define I8 17.346652555743034f  /* 0.7^-8  */
#define I12 72.2476158f         /* 0.7^-12 */

#define XS_STRIDE 20  /* floats per 16-elem chunk row: 80B -> 16B-aligned rows, \
                         16-lane column reads hit 16 distinct LDS banks */

// 0.7^d for d in [0,15], branch-free (4 cndmask+mul), no memory traffic.
__device__ __forceinline__ float pw07(int d) {
  float p = (d & 1) ? 0.7f : 1.0f;
  p *= (d & 2) ? 0.49f : 1.0f;
  p *= (d & 4) ? 0.2401f : 1.0f;
  p *= (d & 8) ? 0.05764801f : 1.0f;
  return p;
}

// ---------------------------------------------------------------------------
// Kernel 1: per-256-element block, local (zero-carry) prefix scan
//   Y = L * X  (L[m,k] = 0.7^{m-k}, X columns = 16-element chunks)
// via 4 chained V_WMMA_F32_16X16X4_F32; intra-block chunk carries with 0.7^16;
// writes W*i into the states channel; emits the exact block-end value.
// One wave per 256-element tile, 8 waves per workgroup. x is staged into LDS
// with GLOBAL_LOAD_ASYNC_TO_LDS_B128 (ASYNCcnt) when the toolchain has it.
// ---------------------------------------------------------------------------
__global__ void alif_syn_scan_wmma(const float* __restrict__ x,
                                   const float* __restrict__ Wsyn,
                                   float* __restrict__ out,
                                   float* __restrict__ blockEnd,
                                   int T, int nTiles) {
  __shared__ float Xs[8][16 * XS_STRIDE];
  __shared__ float Yend[8][16];

  const int lane = threadIdx.x & 31;
  const int wave = threadIdx.x >> 5;
  const int tile = blockIdx.x * 8 + wave;
  const int base = tile * 256;
  const bool active = (tile < nTiles);
  const float w = Wsyn[0];  // uniform scalar load

  if (active) {
    __builtin_prefetch(x + base + 512, 0, 1);  // global_prefetch_b8
#pragma unroll
    for (int h = 0; h < 2; ++h) {
      const int li = h * 128 + lane * 4;  // element within tile, float4 granule
      const int idx = base + li;
      const int c = li >> 4, k = li & 15;  // chunk / pos (float4 stays in row)
      float* lp = &Xs[wave][c * XS_STRIDE + k];
#if __has_builtin(__builtin_amdgcn_global_load_async_to_lds_b128)
      if (idx + 3 < T) {
        // async DMA: 16B global -> LDS, tracked by ASYNCcnt (no VGPR data)
        __builtin_amdgcn_global_load_async_to_lds_b128((gv4p)(x + idx),
                                                       (lv4p)lp, 0, 0);
      } else {
        lp[0] = (idx + 0 < T) ? x[idx + 0] : 0.f;
        lp[1] = (idx + 1 < T) ? x[idx + 1] : 0.f;
        lp[2] = (idx + 2 < T) ? x[idx + 2] : 0.f;
        lp[3] = (idx + 3 < T) ? x[idx + 3] : 0.f;
      }
#else
      if (idx + 3 < T) {
        const float4 xv = *reinterpret_cast<const float4*>(x + idx);
        lp[0] = xv.x; lp[1] = xv.y; lp[2] = xv.z; lp[3] = xv.w;
      } else {
        lp[0] = (idx + 0 < T) ? x[idx + 0] : 0.f;
        lp[1] = (idx + 1 < T) ? x[idx + 1] : 0.f;
        lp[2] = (idx + 2 < T) ? x[idx + 2] : 0.f;
        lp[3] = (idx + 3 < T) ? x[idx + 3] : 0.f;
      }
#endif
    }
  }
#if __has_builtin(__builtin_amdgcn_s_wait_asynccnt)
  __builtin_amdgcn_s_wait_asynccnt(0);  // drain ASYNCcnt before LDS reads
#endif
  __syncthreads();

  const int half = lane >> 4;  // 0: lanes 0-15, 1: lanes 16-31
  const int l16 = lane & 15;   // A: row m; B/D: column (chunk) c
  v8f acc = {0.f, 0.f, 0.f, 0.f, 0.f, 0.f, 0.f, 0.f};

  if (active) {
    // A-fragment powers, branch-free: 0.7^{l16-k0} = pl * 0.7^{-k0}, with
    // 0.7^{-k0} = (const per unrolled g) * (half ? 0.7^-2 : 1). The OOB mask
    // is a single cndmask on the final value -> no exec manipulation.
    const float pl = pw07(l16);             // 0.7^l16, computed once
    const float hInv = half ? I2 : 1.0f;    // 0.7^{-2*half}

    // D[m,c] = sum_k L[m,k] * X[k,c]; K split into 4 groups of 4.
    // A 16x4 layout: lanes 0-15 hold K={k0,k0+1}, lanes 16-31 K={k0+2,k0+3}.
#pragma unroll
    for (int g = 0; g < 4; ++g) {
      const float cg = (g == 0) ? 1.0f : (g == 1) ? I4 : (g == 2) ? I8 : I12;
      const int k0 = 4 * g + 2 * half;
      const int d0 = l16 - k0;
      const float ax = pl * (cg * hInv);  // 0.7^{l16-k0} (when in range)
      v2f a, b;
      a.x = (d0 >= 0) ? ax : 0.f;       // L[m, k0]
      a.y = (d0 >= 1) ? ax * I1 : 0.f;  // L[m, k0+1]
      b.x = Xs[wave][l16 * XS_STRIDE + k0];      // X[k0,   c]
      b.y = Xs[wave][l16 * XS_STRIDE + k0 + 1];  // X[k0+1, c]
      acc = __builtin_amdgcn_wmma_f32_16x16x4_f32(false, a, false, b,
                                                  (short)0, acc, false, false);
    }
    // chunk-end values: Y[15, c] lives in lanes 16-31, component 7
    if (half == 1) Yend[wave][l16] = acc[7];
  }
  __syncthreads();

  if (active) {
    // carry into chunk c = i at end of chunk c-1 (zero block-incoming):
    //   E_c = Yend[c] + 0.7^16 * E_{c-1}
    // fixed trip count + select -> no divergent loop exit, no exec churn
    float E = 0.f;
#pragma unroll
    for (int c = 0; c < 15; ++c) {
      const float En = Yend[wave][c] + Q16 * E;
      E = (c < l16) ? En : E;  // after loop: E = E_{l16-1} (0 for l16==0)
    }

    if (lane == 0) {  // exact block-end (block decay 0.7^256 ~ 2e-40 ~ fp32 0)
      float Eb = 0.f;
#pragma unroll
      for (int c = 0; c < 16; ++c) Eb = Yend[wave][c] + Q16 * Eb;
      blockEnd[tile] = w * Eb;
    }

    const float ph = half ? P8 : 1.0f;  // 0.7^{8*half}
#pragma unroll
    for (int v = 0; v < 8; ++v) {
      const int m = v + 8 * half;         // position within chunk
      const int t = base + l16 * 16 + m;  // global timestep
      // 0.7^{m+1} = pw07(v+1) (compile-time literal after unroll) * ph
      if (t < T) out[T + 4 * t] = w * fmaf(pw07(v + 1) * ph, E, acc[v]);
    }
  }
}

// ---------------------------------------------------------------------------
// Kernel 2: add the cross-block carry. carry_in(block b) = blockEnd[b-1]
// (exact: 0.7^256 underflows fp32); block 0 uses the initial state i0.
//   i[t] += 0.7^{(t%256)+1} * carry_in
// ---------------------------------------------------------------------------
__global__ void alif_syn_carry_fix(float* __restrict__ out,
                                   const float* __restrict__ blockEnd,
                                   const float* __restrict__ state0, int T) {
  const int t = blockIdx.x * blockDim.x + threadIdx.x;
  if (t >= T) return;
  const int b = t >> 8;
  const int j = t & 255;
  const float carry = (b == 0) ? state0[0] : blockEnd[b - 1];
  if (carry != 0.f) {
    const float p = __builtin_exp2f((float)(j + 1) * LOG2_07);  // 0.7^{j+1}
    out[T + 4 * t] = fmaf(p, carry, out[T + 4 * t]);
  }
}

// ---------------------------------------------------------------------------
// Kernel 3: nonlinear membrane/threshold/spike recurrence, one thread per
// 256-step chunk with a 96-step warm-up (0.6^96 ~ 5e-22 << fp32 ulp, so the
// speculative state converges to the serial trajectory before the chunk).
// Reads the exact i from the states channel; writes outputs + v/thr/s.
// ---------------------------------------------------------------------------
__global__ void alif_neuron_scan(float* __restrict__ out,
                                 const float* __restrict__ state0,
                                 int T, int nChunks) {
  const int chunk = blockIdx.x * blockDim.x + threadIdx.x;
  if (chunk >= nChunks) return;
  const int t0 = chunk << 8;
  const int tEnd = (t0 + 256 < T) ? (t0 + 256) : T;
  const float* __restrict__ iarr = out + T;  // i at iarr[4*t]

  float v, thr, s;
  int t;
  if (chunk == 0) {
    v = state0[1]; thr = state0[2]; s = state0[3]; t = 0;
  } else {
    v = 0.f; thr = 0.f; s = 0.f; t = t0 - WARMUP;
  }
  for (; t < t0; ++t) {  // warm-up: advance state, no writes
    const float iv = iarr[4 * t];
    v = LEAK_V * v * (1.f - s) + iv;
    thr = LEAK_T * thr + ADD_T * s;
    s = (v - (BASE_T + thr)) > 0.f ? 1.f : 0.f;
  }
  for (; t < tEnd; ++t) {
    __builtin_prefetch(iarr + 4 * (t + 32), 0, 1);  // L2-resident stream ahead
    const float iv = iarr[4 * t];
    v = LEAK_V * v * (1.f - s) + iv;
    thr = LEAK_T * thr + ADD_T * s;
    s = (v - (BASE_T + thr)) > 0.f ? 1.f : 0.f;
    out[t] = s;                   // outputs[t]
    float* st = out + T + 4 * t;  // states[t] = (i, v, thr, s)
    st[1] = v;
    st[2] = thr;
    st[3] = s;
  }
}

// ---------------------------------------------------------------------------
extern "C" void kernel_launch(void* const* d_in, const int* in_sizes, int n_in,
                              void* d_out, int out_size, void* d_ws,
                              size_t ws_size, hipStream_t stream) {
  const float* x = (const float*)d_in[0];      // [T]
  const float* state = (const float*)d_in[1];  // [4]  (i, v, thr, s)
  const float* Wsyn = (const float*)d_in[2];   // [1]
  float* out = (float*)d_out;                  // [T + 4T]
  const int T = in_sizes[0];
  const int nTiles = (T + 255) / 256;          // 15625 for T = 4e6
  float* blockEnd = (float*)d_ws;              // nTiles floats

  alif_syn_scan_wmma<<<(nTiles + 7) / 8, 256, 0, stream>>>(x, Wsyn, out,
                                                           blockEnd, T, nTiles);
  alif_syn_carry_fix<<<(T + 255) / 256, 256, 0, stream>>>(out, blockEnd, state,
                                                          T);
  alif_neuron_scan<<<(nTiles + 255) / 256, 256, 0, stream>>>(out, state, T,
                                                             nTiles);
}